// HyperConv2dDepthWise_14516989460710
// MI455X (gfx1250) — compile-verified
//
#include <hip/hip_runtime.h>
#include <hip/hip_bf16.h>
#include <stdint.h>

// Problem constants (from reference)
#define BATCH   16
#define CHANS   128
#define IMG_H   128
#define IMG_W   128
#define KSZ     5
#define NWPERB  (KSZ*KSZ*CHANS + CHANS)   // 3328 weights per batch

// Tiling
#define TILE_ROWS 64                // output rows per block (2 row-blocks per plane)
#define LDS_H     (TILE_ROWS + 4)   // 68 rows incl. 2-row halo top/bottom
#define LDS_W     136               // padded row stride in floats (16B-aligned b128 stores)
// valid data lives at LDS x = col + 4  (col in [0,128)), taps read x in [2,133]

typedef float v2f __attribute__((ext_vector_type(2)));

__global__ __launch_bounds__(256)
void HyperConv2dDepthWise_14516989460710_kernel(const float* __restrict__ in,
                                                const float* __restrict__ wts,
                                                float* __restrict__ out)
{
    __shared__ float tile[LDS_H * LDS_W];   // 68*136*4 = 36,992 B

    const int tid    = threadIdx.x;
    const int blk    = blockIdx.x;
    const int rowblk = blk & 1;             // which half of the plane
    const int plane  = blk >> 1;            // b*128 + c
    const int c      = plane & (CHANS - 1);
    const int b      = plane >> 7;
    const int r0     = rowblk * TILE_ROWS;

    const float* __restrict__ planeIn  = in  + (size_t)plane * (IMG_H * IMG_W);
    float*       __restrict__ planeOut = out + (size_t)plane * (IMG_H * IMG_W);

    // ---- block-uniform weights (scalar loads into SGPRs) ----
    const float* __restrict__ wp = wts + (size_t)b * NWPERB + c * (KSZ * KSZ);
    float wk[KSZ * KSZ];
#pragma unroll
    for (int i = 0; i < KSZ * KSZ; ++i) wk[i] = wp[i];
    const float bias = wts[(size_t)b * NWPERB + KSZ * KSZ * CHANS + c];

    // ---- 1) issue async global->LDS copies of 66 valid input rows (b128 per lane) ----
    // rowblk==0: global rows 0..65   -> lds rows 2..67  (lds rows 0,1   zeroed)
    // rowblk==1: global rows 62..127 -> lds rows 0..65  (lds rows 66,67 zeroed)
    const int loadFirst = (rowblk == 0) ? 0 : (r0 - 2);   // first global row loaded
    const int NLOAD     = 66 * 32;                        // 66 rows * 32 x b128 chunks
    for (int i = tid; i < NLOAD; i += 256) {
        const int vr = i >> 5;          // 0..65
        const int cc = i & 31;          // 16-byte chunk within the row
        const int gr = loadFirst + vr;  // global row
        const int lr = gr - (r0 - 2);   // lds row
        // low 32 bits of a generic pointer into LDS == byte offset within wave LDS
        unsigned lds_addr = (unsigned)(uintptr_t)&tile[lr * LDS_W + 4 + cc * 4];
        unsigned long long gaddr =
            (unsigned long long)(uintptr_t)(planeIn + (size_t)gr * IMG_W + cc * 4);
        asm volatile("global_load_async_to_lds_b128 %0, %1, off"
                     :: "v"(lds_addr), "v"(gaddr) : "memory");
    }

    // ---- 2) zero halo cells (disjoint from async-written region) ----
    if (rowblk == 0) {
        for (int j = tid; j < 2 * LDS_W; j += 256) tile[j] = 0.0f;                       // rows 0,1
    } else {
        for (int j = tid; j < 2 * LDS_W; j += 256) tile[(LDS_H - 2) * LDS_W + j] = 0.0f; // rows 66,67
    }
    for (int j = tid; j < LDS_H * 4; j += 256) {            // halo cols 2,3,132,133 each row
        const int r = j >> 2, which = j & 3;
        const int x = (which < 2) ? (2 + which) : (130 + which);
        tile[r * LDS_W + x] = 0.0f;
    }

    // ---- 3) wait for async copies, then sync the workgroup ----
#if __has_builtin(__builtin_amdgcn_s_wait_asynccnt)
    __builtin_amdgcn_s_wait_asynccnt(0);
#else
    asm volatile("s_wait_asynccnt 0" ::: "memory");
#endif
    __syncthreads();

    // ---- 4) packed sliding-window compute, rotating window slots ----
    // Each thread owns one column and 32 rows, processed as TWO parallel 16-row
    // sweeps (rows r and r+16) packed into float2 so every tap is one
    // v_pk_fma_f32 (VOP3P, 2 fp32 FMAs / instruction).
    const int col    = tid & (IMG_W - 1);
    const int rg     = tid >> 7;         // 0 or 1
    const int orBase = rg * 32;          // local output rows [orBase, orBase+32)
    const int xb     = col + 2;          // leftmost tap x index in LDS row

    // splat weights + bias into packed VGPR pairs once (hoisted out of the loop)
    v2f wv[KSZ * KSZ];
#pragma unroll
    for (int i = 0; i < KSZ * KSZ; ++i) wv[i] = (v2f){wk[i], wk[i]};
    const v2f biasPair = (v2f){bias, bias};

    // slot s holds lds rows (orBase + it + d) with s == (it + d) % 5
    v2f win[5][5];   // .x -> sweep A (rows orBase+*), .y -> sweep B (rows orBase+16+*)
#pragma unroll
    for (int r = 0; r < 4; ++r)          // preload rows orBase+0..3 into slots 0..3
#pragma unroll
        for (int x = 0; x < 5; ++x) {
            win[r][x].x = tile[(orBase + r) * LDS_W + xb + x];
            win[r][x].y = tile[(orBase + 16 + r) * LDS_W + xb + x];
        }

#pragma unroll
    for (int it = 0; it < 16; ++it) {
        const int orA     = orBase + it;   // sweep A output row (local)
        const int orB     = orA + 16;      // sweep B output row (local)
        const int slotNew = (it + 4) % 5;  // compile-time after unroll

        // fetch the new bottom rows of both 5x5 windows (10 DS loads)
#pragma unroll
        for (int x = 0; x < 5; ++x) {
            win[slotNew][x].x = tile[(orA + 4) * LDS_W + xb + x];
            win[slotNew][x].y = tile[(orB + 4) * LDS_W + xb + x];
        }

        // two packed accumulators (chain depth ~13); no per-iteration init moves:
        //   tap0: acc0 = win*w + {bias,bias}   (C operand carries the bias)
        //   tap1: acc1 = win*w                 (v_pk_mul defines acc1)
        const int s0 = it % 5;             // compile-time after unroll
        v2f acc0, acc1;
        asm("v_pk_fma_f32 %0, %1, %2, %3"
            : "=v"(acc0) : "v"(win[s0][0]), "v"(wv[0]), "v"(biasPair));
        asm("v_pk_mul_f32 %0, %1, %2"
            : "=v"(acc1) : "v"(win[s0][1]), "v"(wv[1]));
#pragma unroll
        for (int d = 0; d < 5; ++d) {
            const int s = (it + d) % 5;    // compile-time after unroll
#pragma unroll
            for (int x = 0; x < 5; ++x) {
                const int t = d * 5 + x;
                if (t < 2) continue;       // taps 0,1 done above
                v2f& a = (t & 1) ? acc1 : acc0;
                asm("v_pk_fma_f32 %0, %1, %2, %0"
                    : "+v"(a)
                    : "v"(win[s][x]), "v"(wv[t]));
            }
        }
        const float vA = acc0.x + acc1.x;
        const float vB = acc0.y + acc1.y;

        // SiLU on both rows: v * sigmoid(v) via v_exp_f32 + v_rcp_f32
        const float eA = __builtin_amdgcn_exp2f(vA * -1.44269504088896341f);
        const float eB = __builtin_amdgcn_exp2f(vB * -1.44269504088896341f);
        const float oA = vA * __builtin_amdgcn_rcpf(1.0f + eA);
        const float oB = vB * __builtin_amdgcn_rcpf(1.0f + eB);
        planeOut[(size_t)(r0 + orA) * IMG_W + col] = oA;
        planeOut[(size_t)(r0 + orB) * IMG_W + col] = oB;
    }
}

extern "C" void kernel_launch(void* const* d_in, const int* in_sizes, int n_in,
                              void* d_out, int out_size, void* d_ws, size_t ws_size,
                              hipStream_t stream)
{
    (void)in_sizes; (void)n_in; (void)out_size; (void)d_ws; (void)ws_size;
    const float* in  = (const float*)d_in[0];
    const float* wts = (const float*)d_in[1];
    float*       out = (float*)d_out;

    dim3 grid(BATCH * CHANS * 2);   // 4096 blocks: one (b,c) half-plane each
    dim3 block(256);                // 8 wave32 waves
    HyperConv2dDepthWise_14516989460710_kernel<<<grid, block, 0, stream>>>(in, wts, out);
}